// SplineCNN_87024627352318
// MI455X (gfx1250) — compile-verified
//
#include <hip/hip_runtime.h>
#include <hip/hip_bf16.h>

typedef __attribute__((ext_vector_type(2))) float v2f;
typedef __attribute__((ext_vector_type(8))) float v8f;

// ---- problem constants (from reference) ----
constexpr int KK     = 5;        // kernel size per dim
constexpr int SS     = 4;        // 2^DIM contributing knots
constexpr int KPROD  = 25;       // K^DIM
constexpr int NN     = 50000;    // nodes  (divisible by 16 -> full WMMA tiles)
constexpr int EE     = 800000;   // edges
constexpr int IN_C   = 32;
constexpr int OUT_C  = 64;
constexpr int FIN    = IN_C + 2 * OUT_C;   // 160

// ---------------------------------------------------------------------------
// 1) B-spline basis + kernel indices + degree (one thread per edge)
// ---------------------------------------------------------------------------
__global__ void basis_deg_kernel(const long long* __restrict__ edge_index,
                                 const float* __restrict__ edge_attr,
                                 float* __restrict__ basis,
                                 int*   __restrict__ widx,
                                 float* __restrict__ deg) {
  int e = blockIdx.x * blockDim.x + threadIdx.x;
  if (e >= EE) return;
  float u = edge_attr[2 * e + 0] * (float)(KK - 1);
  float v = edge_attr[2 * e + 1] * (float)(KK - 1);
  float bu = floorf(u), bv = floorf(v);
  float fu = u - bu,   fv = v - bv;
  int   iu = (int)bu,  iv = (int)bv;
#pragma unroll
  for (int s = 0; s < SS; ++s) {
    int b0 = s & 1, b1 = (s >> 1) & 1;
    float wgt = (b0 ? fu : 1.0f - fu) * (b1 ? fv : 1.0f - fv);
    int ku = min(max(iu + b0, 0), KK - 1);
    int kv = min(max(iv + b1, 0), KK - 1);
    basis[e * SS + s] = wgt;
    widx [e * SS + s] = ku + KK * kv;          // strides = [1, K]
  }
  int dst = (int)edge_index[EE + e];
  atomicAdd(&deg[dst], 1.0f);
}

// deg -> 1/max(deg,1)  (in place, recomputed every call -> deterministic)
__global__ void rdeg_kernel(float* __restrict__ deg) {
  int n = blockIdx.x * blockDim.x + threadIdx.x;
  if (n >= NN) return;
  float d = deg[n];
  deg[n] = 1.0f / (d > 1.0f ? d : 1.0f);
}

// ---------------------------------------------------------------------------
// 2) Scatter basis-weighted source features: acc[dst, k_s, :] += b_s * x[src, :]
//    ONE WAVE PER EDGE: x[src] loaded once, reused for all 4 knots.
// ---------------------------------------------------------------------------
template <int CIN>
__global__ void scatter_kernel(const long long* __restrict__ edge_index,
                               const float* __restrict__ basis,
                               const int*   __restrict__ widx,
                               const float* __restrict__ xin,
                               float* __restrict__ acc) {
  int gtid = blockIdx.x * blockDim.x + threadIdx.x;
  int e    = gtid >> 5;
  int lane = gtid & 31;
  if (e >= EE) return;
  int src = (int)edge_index[e];
  int dst = (int)edge_index[EE + e];

  float xv0 = xin[(size_t)src * CIN + lane];
  float xv1 = 0.0f;
  if constexpr (CIN > 32) xv1 = xin[(size_t)src * CIN + 32 + lane];

  float* base = acc + (size_t)dst * KPROD * CIN;
#pragma unroll
  for (int s = 0; s < SS; ++s) {
    float wgt = basis[e * SS + s];
    int   k   = widx [e * SS + s];
    float* ap = base + (size_t)k * CIN;
    atomicAdd(&ap[lane], wgt * xv0);
    if constexpr (CIN > 32) atomicAdd(&ap[32 + lane], wgt * xv1);
  }
}

// ---------------------------------------------------------------------------
// 3) SplineConv dense stage (WMMA f32 16x16x4), A-tile staged in LDS:
//    out[m,:] = relu( (sum_k acc[m,k,:] @ w[k]) * rdeg[m] + x[m,:]@root + b )
//    block = 128 thr = 4 waves; wave 'w' owns output columns [16w, 16w+16)
//    LDS: contiguous 16-node slab of acc (16*KPROD*CIN floats), loaded once,
//    consumed by all 4 waves (A is identical across waves).
// ---------------------------------------------------------------------------
template <int CIN>
__global__ void spline_gemm_kernel(const float* __restrict__ xin,
                                   const float* __restrict__ acc,
                                   const float* __restrict__ w,
                                   const float* __restrict__ root,
                                   const float* __restrict__ bias,
                                   const float* __restrict__ rdeg,
                                   float* __restrict__ xout) {
  extern __shared__ float sA[];            // 16 * KPROD * CIN floats
  const int tid  = threadIdx.x;
  const int lane = tid & 31;
  const int wave = tid >> 5;               // 0..3 -> output col tile
  const int m0   = blockIdx.x * 16;
  const int half = lane >> 4;              // 0: K=0,1 / M=0..7 ; 1: K=2,3 / M=8..15
  const int r    = lane & 15;
  const int col  = wave * 16 + r;

  // cooperative, fully-coalesced slab copy: acc[m0..m0+15] is contiguous
  {
    const float4* g4 = (const float4*)(acc + (size_t)m0 * KPROD * CIN);
    float4*       s4 = (float4*)sA;
    const int total4 = 16 * KPROD * CIN / 4;
    for (int i = tid; i < total4; i += 128) s4[i] = g4[i];
  }
  __syncthreads();

  v8f cagg = {};
  const float* arow = sA + r * KPROD * CIN + 2 * half;   // LDS A fragments
  for (int k = 0; k < KPROD; ++k) {
    const float* ak = arow + k * CIN;
    const float* bk = w + (size_t)k * CIN * OUT_C + col;
#pragma unroll
    for (int kk = 0; kk < CIN / 4; ++kk) {
      v2f a; a.x = ak[kk * 4];                       a.y = ak[kk * 4 + 1];
      v2f b; b.x = bk[(kk * 4 + 2 * half) * OUT_C];  b.y = bk[(kk * 4 + 2 * half + 1) * OUT_C];
      cagg = __builtin_amdgcn_wmma_f32_16x16x4_f32(false, a, false, b,
                                                   (short)0, cagg, false, false);
    }
  }

  v8f croot = {};
  {
    const float* ax = xin + (size_t)(m0 + r) * CIN + 2 * half;
    const float* br = root + col;
#pragma unroll
    for (int kk = 0; kk < CIN / 4; ++kk) {
      v2f a; a.x = ax[kk * 4];                       a.y = ax[kk * 4 + 1];
      v2f b; b.x = br[(kk * 4 + 2 * half) * OUT_C];  b.y = br[(kk * 4 + 2 * half + 1) * OUT_C];
      croot = __builtin_amdgcn_wmma_f32_16x16x4_f32(false, a, false, b,
                                                    (short)0, croot, false, false);
    }
  }

  const float bv = bias[col];
#pragma unroll
  for (int j = 0; j < 8; ++j) {
    int m = m0 + j + 8 * half;
    float v = cagg[j] * rdeg[m] + croot[j] + bv;
    xout[(size_t)m * OUT_C + col] = v > 0.0f ? v : 0.0f;
  }
}

// ---------------------------------------------------------------------------
// 4) Final linear over the virtual concat [x | x1 | x2] (FIN=160) @ fw + fb
// ---------------------------------------------------------------------------
__device__ __forceinline__ float h_elem(const float* x0, const float* x1,
                                        const float* x2, int m, int t) {
  if (t < IN_C)            return x0[(size_t)m * IN_C + t];
  if (t < IN_C + OUT_C)    return x1[(size_t)m * OUT_C + (t - IN_C)];
  return x2[(size_t)m * OUT_C + (t - IN_C - OUT_C)];
}

__global__ void final_gemm_kernel(const float* __restrict__ x0,
                                  const float* __restrict__ x1,
                                  const float* __restrict__ x2,
                                  const float* __restrict__ fw,
                                  const float* __restrict__ fb,
                                  float* __restrict__ out) {
  const int lane = threadIdx.x & 31;
  const int wave = threadIdx.x >> 5;
  const int m0   = blockIdx.x * 16;
  const int half = lane >> 4;
  const int r    = lane & 15;
  const int col  = wave * 16 + r;

  v8f c = {};
#pragma unroll
  for (int kk = 0; kk < FIN / 4; ++kk) {
    int t0 = kk * 4 + 2 * half;
    v2f a; a.x = h_elem(x0, x1, x2, m0 + r, t0);
           a.y = h_elem(x0, x1, x2, m0 + r, t0 + 1);
    v2f b; b.x = fw[(size_t)t0 * OUT_C + col];
           b.y = fw[(size_t)(t0 + 1) * OUT_C + col];
    c = __builtin_amdgcn_wmma_f32_16x16x4_f32(false, a, false, b,
                                              (short)0, c, false, false);
  }
  const float bv = fb[col];
#pragma unroll
  for (int j = 0; j < 8; ++j) {
    int m = m0 + j + 8 * half;
    out[(size_t)m * OUT_C + col] = c[j] + bv;
  }
}

// ---------------------------------------------------------------------------
extern "C" void kernel_launch(void* const* d_in, const int* in_sizes, int n_in,
                              void* d_out, int out_size, void* d_ws, size_t ws_size,
                              hipStream_t stream) {
  (void)in_sizes; (void)n_in; (void)out_size; (void)ws_size;

  const float*     x     = (const float*)    d_in[0];
  const long long* ei    = (const long long*)d_in[1];   // int64 edge_index [2,E]
  const float*     ea    = (const float*)    d_in[2];
  const float*     w0    = (const float*)    d_in[3];
  const float*     root0 = (const float*)    d_in[4];
  const float*     b0    = (const float*)    d_in[5];
  const float*     w1    = (const float*)    d_in[6];
  const float*     root1 = (const float*)    d_in[7];
  const float*     b1    = (const float*)    d_in[8];
  const float*     fw    = (const float*)    d_in[9];
  const float*     fb    = (const float*)    d_in[10];
  float*           out   = (float*)d_out;

  // ---- workspace carve-out (256B aligned regions) ----
  char* p = (char*)d_ws;
  auto carve = [&](size_t bytes) {
    char* q = p;
    p += (bytes + 255) & ~(size_t)255;
    return q;
  };
  float* basis = (float*)carve((size_t)EE * SS * sizeof(float));   // 12.8 MB
  int*   widx  = (int*)  carve((size_t)EE * SS * sizeof(int));     // 12.8 MB
  float* deg   = (float*)carve((size_t)NN * sizeof(float));        //  0.2 MB
  float* x1b   = (float*)carve((size_t)NN * OUT_C * sizeof(float));// 12.8 MB
  float* x2b   = (float*)carve((size_t)NN * OUT_C * sizeof(float));// 12.8 MB
  float* acc   = (float*)carve((size_t)NN * KPROD * OUT_C * sizeof(float)); // 320 MB

  // ---- basis / degree / reciprocal-degree ----
  hipMemsetAsync(deg, 0, (size_t)NN * sizeof(float), stream);
  basis_deg_kernel<<<(EE + 255) / 256, 256, 0, stream>>>(ei, ea, basis, widx, deg);
  rdeg_kernel<<<(NN + 255) / 256, 256, 0, stream>>>(deg);

  const int scatter_blocks = (EE * 32 + 255) / 256;   // one wave per edge

  // ---- layer 0: cin = 32 ----
  hipMemsetAsync(acc, 0, (size_t)NN * KPROD * IN_C * sizeof(float), stream);
  scatter_kernel<IN_C><<<scatter_blocks, 256, 0, stream>>>(ei, basis, widx, x, acc);
  spline_gemm_kernel<IN_C>
      <<<NN / 16, 128, 16 * KPROD * IN_C * sizeof(float), stream>>>(
          x, acc, w0, root0, b0, deg, x1b);

  // ---- layer 1: cin = 64 ----
  hipMemsetAsync(acc, 0, (size_t)NN * KPROD * OUT_C * sizeof(float), stream);
  scatter_kernel<OUT_C><<<scatter_blocks, 256, 0, stream>>>(ei, basis, widx, x1b, acc);
  spline_gemm_kernel<OUT_C>
      <<<NN / 16, 128, 16 * KPROD * OUT_C * sizeof(float), stream>>>(
          x1b, acc, w1, root1, b1, deg, x2b);

  // ---- final linear on concat(x, x1, x2) ----
  final_gemm_kernel<<<NN / 16, 128, 0, stream>>>(x, x1b, x2b, fw, fb, out);
}